// SetOfOrthonormalTransforms_44933947851298
// MI455X (gfx1250) — compile-verified
//
#include <hip/hip_runtime.h>
#include <hip/hip_bf16.h>

// SetOfOrthonormalTransforms on MI455X (gfx1250).
// out[b] = (mus[b] * Givens(angles[b])) @ X[b],  b in [0,4096), 8x8 @ 8x1024, fp32.
//
// Memory-bound (~2 FLOP/byte, ~256MB traffic @ 23.3TB/s). Strategy:
//  * Phase 1: build R per block in registers (one column per lane, 28 unrolled
//    Givens steps), scale rows by mus, stage to LDS.
//  * Phase 2: one wave per pair of consecutive blocks. A = diag(R_b0,R_b1)
//    (16x16 block diagonal) held as 4 x v2f A-fragments in the ISA's
//    16x4-f32 A layout. Stream X in 16-column tiles; K=16 via 4 chained
//    V_WMMA_F32_16X16X4_F32; store D per the documented 16x16 f32 C/D layout.
//    Consecutive blocks are contiguous, so the "stacked" 16xL operand is just
//    16 contiguous rows of X -> perfectly coalesced loads/stores.

typedef __attribute__((ext_vector_type(2))) float v2f;
typedef __attribute__((ext_vector_type(8))) float v8f;

#define NPTS 8
#define NANG 28
#define LCOLS 1024
#define BLKS_PER_WG 16   // 8 waves/WG * 2 blocks/wave

__global__ __launch_bounds__(256)
void SetOfOrthonormalTransforms_44933947851298_kernel(
    const float* __restrict__ X,
    const float* __restrict__ angles,
    const float* __restrict__ mus,
    float* __restrict__ out)
{
    __shared__ float Rlds[BLKS_PER_WG][NPTS * NPTS];  // row-major 8x8 per block

    const int tid    = threadIdx.x;
    const int wgBlk0 = blockIdx.x * BLKS_PER_WG;

    // ---------- Phase 1: build R for this WG's 16 blocks ----------
    // Threads 0..127: thread (wb, col) owns column `col` of block wb's R.
    // Givens update acts identically on every column element:
    //   m[i]' = c*m[i] - s*m[j];  m[j]' = s*m[i] + c*m[j]
    if (tid < BLKS_PER_WG * NPTS) {
        const int wb  = tid >> 3;   // local block 0..15
        const int col = tid & 7;
        const int b   = wgBlk0 + wb;
        const float* __restrict__ ang = angles + (size_t)b * NANG;

        float m[NPTS];
        #pragma unroll
        for (int i = 0; i < NPTS; ++i) m[i] = (i == col) ? 1.0f : 0.0f;

        int k = 0;
        #pragma unroll
        for (int i = 0; i < NPTS - 1; ++i) {
            #pragma unroll
            for (int j = i + 1; j < NPTS; ++j) {
                const float a = ang[k++];
                float s, c;
                __sincosf(a, &s, &c);
                const float vt = m[i], vb = m[j];
                m[i] = c * vt - s * vb;
                m[j] = s * vt + c * vb;
            }
        }
        const float* __restrict__ mu = mus + (size_t)b * NPTS;
        #pragma unroll
        for (int i = 0; i < NPTS; ++i)
            Rlds[wb][i * NPTS + col] = m[i] * mu[i];
    }
    __syncthreads();

    // ---------- Phase 2: WMMA stream. EXEC is all-ones here. ----------
    const int wave = tid >> 5;        // 0..7
    const int lane = tid & 31;
    const int m16  = lane & 15;       // A-matrix row M (0..15)
    const int half = lane >> 4;       // 0 -> K%4 in {0,1}; 1 -> {2,3}

    // Build A = diag(R_b0, R_b1) fragments, 16x4 f32 A layout per chunk:
    //   lane 0-15 : VGPR0 = A[M][4k+0], VGPR1 = A[M][4k+1]
    //   lane16-31 : VGPR0 = A[M][4k+2], VGPR1 = A[M][4k+3]
    const int bl   = (wave << 1) | (m16 >> 3);   // local block owning row M
    const int row  = m16 & 7;
    const float* __restrict__ Rrow = &Rlds[bl][row * NPTS];

    v2f A[4];
    #pragma unroll
    for (int kk = 0; kk < 4; ++kk) {
        const int k0 = kk * 4 + half * 2;
        const int k1 = k0 + 1;
        v2f a;
        a.x = ((k0 >> 3) == (m16 >> 3)) ? Rrow[k0 & 7] : 0.0f;
        a.y = ((k1 >> 3) == (m16 >> 3)) ? Rrow[k1 & 7] : 0.0f;
        A[kk] = a;
    }

    // Two consecutive blocks -> stacked 16xL operand is 16 contiguous rows.
    const size_t base = (size_t)(wgBlk0 + wave * 2) * NPTS * LCOLS;
    const float* __restrict__ Xb = X   + base;
    float*       __restrict__ Ob = out + base;

    #pragma unroll 2
    for (int c0 = 0; c0 < LCOLS; c0 += 16) {
        v8f acc = {};
        #pragma unroll
        for (int kk = 0; kk < 4; ++kk) {
            // B 4x16 f32 chunk: lane = column c0+m16;
            //   VGPR0 = X[row kA], VGPR1 = X[row kA+1]  (kA = 4kk + 2*half)
            const int kA = kk * 4 + half * 2;
            const float* __restrict__ p = Xb + (size_t)kA * LCOLS + c0 + m16;
            v2f B;
            B.x = p[0];
            B.y = p[LCOLS];
            acc = __builtin_amdgcn_wmma_f32_16x16x4_f32(
                /*neg_a=*/false, A[kk], /*neg_b=*/false, B,
                /*c_mod=*/(short)0, acc, /*reuse_a=*/false, /*reuse_b=*/false);
        }
        // D 16x16 f32 layout: VGPR v -> row v (lanes 0-15) or v+8 (lanes 16-31)
        #pragma unroll
        for (int v = 0; v < 8; ++v) {
            Ob[(size_t)(v + half * 8) * LCOLS + c0 + m16] = acc[v];
        }
    }
}

extern "C" void kernel_launch(void* const* d_in, const int* in_sizes, int n_in,
                              void* d_out, int out_size, void* d_ws, size_t ws_size,
                              hipStream_t stream) {
    const float* X      = (const float*)d_in[0];
    const float* angles = (const float*)d_in[1];
    const float* mus    = (const float*)d_in[2];
    float*       out    = (float*)d_out;

    const int nblks = in_sizes[2] / NPTS;        // mus is [nblks, 8]
    const int grid  = nblks / BLKS_PER_WG;       // 4096/16 = 256 workgroups

    SetOfOrthonormalTransforms_44933947851298_kernel<<<grid, 256, 0, stream>>>(
        X, angles, mus, out);
}